// StepConditionalLoReFT_78872779423981
// MI455X (gfx1250) — compile-verified
//
#include <hip/hip_runtime.h>
#include <math.h>

// Problem constants (from reference): h (B=8, S=50, T=77, D=1024); R,W (S,16,D); b (S,16)
#define NB 8
#define NS 50
#define NT 77
#define ND 1024
#define NR 16
#define TTILE 16
#define NTILES 5          // ceil(77/16)
#define HSTR 1028         // padded LDS row stride in floats (16B-aligned, bank-staggered)
#define PSTR 17           // padded Pm stride (17 coprime 64 -> conflict-free)

typedef __attribute__((ext_vector_type(2))) float v2f;
typedef __attribute__((ext_vector_type(4))) float v4f;
typedef __attribute__((ext_vector_type(8))) float v8f;

// ---------------------------------------------------------------------------
// Kernel 1: inv_norm[s*16 + r] = 1 / max(||R[s,r,:]||, 1e-12)
// ---------------------------------------------------------------------------
__global__ __launch_bounds__(256)
void loreft_rownorm_kernel(const float* __restrict__ R, float* __restrict__ inv)
{
    const int sr = blockIdx.x;                 // 0 .. NS*NR-1
    const float* row = R + (size_t)sr * ND;

    float sum = 0.0f;
    for (int i = threadIdx.x; i < ND; i += 256) {
        float v = row[i];
        sum += v * v;
    }
    __shared__ float red[256];
    red[threadIdx.x] = sum;
    __syncthreads();
    for (int off = 128; off > 0; off >>= 1) {
        if (threadIdx.x < off) red[threadIdx.x] += red[threadIdx.x + off];
        __syncthreads();
    }
    if (threadIdx.x == 0) {
        float n = sqrtf(red[0]);
        n = fmaxf(n, 1e-12f);
        inv[sr] = 1.0f / n;
    }
}

// ---------------------------------------------------------------------------
// Kernel 2: fused LoReFT edit. Block = 4 wave32s, one (b, s, 16-token tile).
//   Stage 0: coalesced global->LDS copy of the 16x1024 h tile (single HBM read).
//   Phase A: Rh,Wh via v_wmma_f32_16x16x4_f32, A-fragments fed from LDS,
//            K=D=1024 split 4 ways across waves.
//   LDS reduction -> P = Wh - Rh + b (16x16, stride-17 padded).
//   Phase B: delta = P x Rn (K=r=16), residual h re-read from LDS, store out.
// ---------------------------------------------------------------------------
__global__ __launch_bounds__(128)
void loreft_main_kernel(const float* __restrict__ h,
                        const float* __restrict__ R,
                        const float* __restrict__ W,
                        const float* __restrict__ bvec,
                        const float* __restrict__ inv,
                        float* __restrict__ out)
{
    extern __shared__ float smem[];
    float* __restrict__ htile   = smem;                        // TTILE * HSTR
    float* __restrict__ partial = htile + TTILE * HSTR;        // 4 * 16 * 16
    float* __restrict__ Pm      = partial + 4 * TTILE * NR;    // 16 * PSTR
    float* __restrict__ invL    = Pm + TTILE * PSTR;           // 16
    float* __restrict__ bL      = invL + NR;                   // 16

    const int blk  = blockIdx.x;
    const int tile = blk % NTILES;
    const int bs   = blk / NTILES;
    const int s    = bs % NS;
    const int b    = bs / NS;
    const int t0   = tile * TTILE;

    const int lane = threadIdx.x & 31;
    const int wave = threadIdx.x >> 5;      // 0..3
    const int m    = lane & 15;             // row (A) / col (B) index within 16
    const int kk   = (lane >> 4) << 1;      // 0 or 2: K sub-offset within 4-wide chunk

    const size_t tok_base = ((size_t)(b * NS + s)) * NT;
    const float* __restrict__ hbase = h + tok_base * ND;
    float*       __restrict__ obase = out + tok_base * ND;
    const float* __restrict__ Rs = R + (size_t)s * NR * ND;
    const float* __restrict__ Ws = W + (size_t)s * NR * ND;

    // Per-step scalars into LDS.
    if (threadIdx.x < NR) {
        invL[threadIdx.x] = inv[s * NR + threadIdx.x];
        bL[threadIdx.x]   = bvec[s * NR + threadIdx.x];
    }

    // ---- Stage 0: coalesced h tile -> LDS (the only HBM read of h) --------
    // 16 rows * 256 float4 = 4096 float4; 128 threads -> 32 iterations.
    for (int idx = threadIdx.x; idx < TTILE * (ND / 4); idx += 128) {
        int row  = idx >> 8;                 // / 256
        int c4   = idx & 255;                // float4 within row
        int tokr = t0 + row;
        tokr = (tokr < NT) ? tokr : (NT - 1);   // clamp ragged tail; masked at store
        v4f v = *(const v4f*)(hbase + (size_t)tokr * ND + ((size_t)c4 << 2));
        *(v4f*)(htile + row * HSTR + (c4 << 2)) = v;
    }
    __syncthreads();

    // ---------------- Phase A: Rh, Wh (16x16 partials per wave) -------------
    const float invm = invL[m];
    const float* __restrict__ Rrow = Rs + (size_t)m * ND;   // B lane owns r-row m
    const float* __restrict__ Wrow = Ws + (size_t)m * ND;
    const float* __restrict__ hL   = htile + m * HSTR;      // LDS row of this lane's token

    v8f accR = {};
    v8f accW = {};
    const int k0 = wave * (ND / 4);           // 256-wide K slice per wave
    for (int k = k0; k < k0 + ND / 4; k += 4) {
        v2f a  = *(const v2f*)(hL + k + kk);       // ds_load_b64, conflict-free
        v2f br = *(const v2f*)(Rrow + k + kk);     // L2-resident per-step weights
        v2f bw = *(const v2f*)(Wrow + k + kk);
        br.x *= invm;
        br.y *= invm;
        accR = __builtin_amdgcn_wmma_f32_16x16x4_f32(
                   false, a, false, br, (short)0, accR, false, false);
        accW = __builtin_amdgcn_wmma_f32_16x16x4_f32(
                   false, a, false, bw, (short)0, accW, false, false);
    }

    // ------------- Cross-wave reduction: P = Wh - Rh + b (16x16) ------------
    #pragma unroll
    for (int i = 0; i < 8; ++i) {
        int row = i + ((lane >> 4) << 3);   // C/D layout: lanes 16-31 hold M+8
        int col = lane & 15;
        partial[(wave * TTILE + row) * NR + col] = accW[i] - accR[i];
    }
    __syncthreads();
    for (int idx = threadIdx.x; idx < TTILE * NR; idx += 128) {
        int row = idx >> 4, col = idx & 15;
        Pm[row * PSTR + col] =
              partial[(0 * TTILE + row) * NR + col]
            + partial[(1 * TTILE + row) * NR + col]
            + partial[(2 * TTILE + row) * NR + col]
            + partial[(3 * TTILE + row) * NR + col]
            + bL[col];
    }
    __syncthreads();

    // ---------------- Phase B: delta = P x Rn, fused residual ---------------
    // M=16 tokens, K=16 (r), N = 16-wide D tiles; 64 tiles split 4 ways.
    for (int j = 0; j < 16; ++j) {
        const int d0   = ((j << 2) | wave) << 4;   // (j*4 + wave) * 16
        const int dcol = d0 + m;                   // B lane owns D column dcol
        v8f acc = {};
        #pragma unroll
        for (int kc = 0; kc < NR; kc += 4) {
            v2f a, br;
            a.x  = Pm[m * PSTR + kc + kk];
            a.y  = Pm[m * PSTR + kc + kk + 1];
            br.x = Rs[(size_t)(kc + kk)     * ND + dcol] * invL[kc + kk];
            br.y = Rs[(size_t)(kc + kk + 1) * ND + dcol] * invL[kc + kk + 1];
            acc = __builtin_amdgcn_wmma_f32_16x16x4_f32(
                      false, a, false, br, (short)0, acc, false, false);
        }
        #pragma unroll
        for (int i = 0; i < 8; ++i) {
            int row  = i + ((lane >> 4) << 3);
            int toko = t0 + row;
            if (toko < NT) {
                int col = d0 + (lane & 15);
                obase[(size_t)toko * ND + col] =
                    htile[row * HSTR + col] + acc[i];   // residual from LDS, not HBM
            }
        }
    }
}

// ---------------------------------------------------------------------------
extern "C" void kernel_launch(void* const* d_in, const int* in_sizes, int n_in,
                              void* d_out, int out_size, void* d_ws, size_t ws_size,
                              hipStream_t stream)
{
    const float* h    = (const float*)d_in[0];   // (8,50,77,1024)
    const float* R    = (const float*)d_in[1];   // (50,16,1024)
    const float* W    = (const float*)d_in[2];   // (50,16,1024)
    const float* bvec = (const float*)d_in[3];   // (50,16)
    float* out = (float*)d_out;
    float* inv = (float*)d_ws;                   // 800 floats of scratch

    (void)in_sizes; (void)n_in; (void)out_size; (void)ws_size;

    enum { SMEM_FLOATS = TTILE * HSTR + 4 * TTILE * NR + TTILE * PSTR + NR + NR };

    loreft_rownorm_kernel<<<NS * NR, 256, 0, stream>>>(R, inv);
    loreft_main_kernel<<<NB * NS * NTILES, 128, SMEM_FLOATS * sizeof(float), stream>>>(
        h, R, W, bvec, inv, out);
}